// GATLayer_14499809591803
// MI455X (gfx1250) — compile-verified
//
#include <hip/hip_runtime.h>
#include <math.h>

// GAT forward for MI455X (gfx1250, wave32).
// N=50000, E=800000, IN_F=256, OUT_F=64, NHEAD=4, ALPHA=0.2
//
// Pipeline:
//  K0 init      : out = 0, emax = -inf, denom = 0
//  K1 gemm      : h = x @ W via V_WMMA_F32_16X16X4_F32 (wave computes 16x64 slab)
//  K2 logits    : hl[n,h] = sum_f a_l[h,f]*h[n,h,f]; hr likewise (wave per node)
//  K3 edge max  : e = lrelu(hl[row]+hr[col]); atomic segment-max per (row,head)
//  K4 edge exp  : ee = exp(e - emax[row]); atomic segment-sum denom
//  K5 aggregate : out[row] += (ee/denom) * h[col]   (wave per edge, f32 atomics, L2-resident)

typedef float v2f __attribute__((ext_vector_type(2)));
typedef float v8f __attribute__((ext_vector_type(8)));

#define IN_F   256
#define OUTF   64
#define NHEAD  4
#define TOTF   256   // NHEAD*OUTF
#define ALPHA  0.2f

__device__ __forceinline__ void atomicMaxF32(float* addr, float val) {
  unsigned int* uaddr = (unsigned int*)addr;
  unsigned int old = *uaddr;
  while (__uint_as_float(old) < val) {
    unsigned int assumed = old;
    old = atomicCAS(uaddr, assumed, __float_as_uint(val));
    if (old == assumed) break;
  }
}

// ---------------- K0: init ----------------
__global__ void gat_init(float* __restrict__ out, float* __restrict__ emax,
                         float* __restrict__ denom, int n_out, int n_seg) {
  int i = blockIdx.x * blockDim.x + threadIdx.x;
  if (i < n_out) out[i] = 0.0f;
  if (i < n_seg) { emax[i] = -__builtin_huge_valf(); denom[i] = 0.0f; }
}

// ---------------- K1: h = x @ W via f32 WMMA ----------------
// Block = 128 threads = 4 waves. Block handles 16 rows x 256 cols.
// Wave w handles cols [64w, 64w+64): one A-frag reused across 4 WMMAs.
__global__ void __launch_bounds__(128)
gat_gemm_wmma(const float* __restrict__ x, const float* __restrict__ W,
              float* __restrict__ h, int N) {
  const int lane    = threadIdx.x & 31;
  const int wave    = threadIdx.x >> 5;      // 0..3
  const int rowBase = blockIdx.x * 16;
  const int colBase = wave * 64;
  const int l15     = lane & 15;
  const int hi      = lane >> 4;             // 0 or 1 (K-pair select)

  int mrow = rowBase + l15;
  if (mrow >= N) mrow = N - 1;               // N=50000 is a multiple of 16; safety clamp
  const float* __restrict__ xrow = x + (size_t)mrow * IN_F;

  v8f acc0 = {}, acc1 = {}, acc2 = {}, acc3 = {};
  for (int kb = 0; kb < IN_F; kb += 4) {
    const int k0 = kb + 2 * hi;
    v2f a;                                    // A 16x4: lane holds (M=l15, K=k0..k0+1)
    a.x = xrow[k0];
    a.y = xrow[k0 + 1];
    const float* __restrict__ w0 = W + (size_t)k0 * TOTF + colBase + l15;
    const float* __restrict__ w1 = w0 + TOTF;
    v2f b;                                    // B 4x16: lane holds (K=k0..k0+1, N=l15)
    b.x = w0[0];  b.y = w1[0];
    acc0 = __builtin_amdgcn_wmma_f32_16x16x4_f32(false, a, false, b, (short)0, acc0, false, false);
    b.x = w0[16]; b.y = w1[16];
    acc1 = __builtin_amdgcn_wmma_f32_16x16x4_f32(false, a, false, b, (short)0, acc1, false, false);
    b.x = w0[32]; b.y = w1[32];
    acc2 = __builtin_amdgcn_wmma_f32_16x16x4_f32(false, a, false, b, (short)0, acc2, false, false);
    b.x = w0[48]; b.y = w1[48];
    acc3 = __builtin_amdgcn_wmma_f32_16x16x4_f32(false, a, false, b, (short)0, acc3, false, false);
  }

  // D layout: VGPR r -> (M = r + 8*hi, N = l15)
  const int srow = rowBase + 8 * hi;
#pragma unroll
  for (int r = 0; r < 8; ++r) {
    const int gr = srow + r;
    if (gr < N) {
      float* hp = h + (size_t)gr * TOTF + colBase + l15;
      hp[0]  = acc0[r];
      hp[16] = acc1[r];
      hp[32] = acc2[r];
      hp[48] = acc3[r];
    }
  }
}

// ---------------- K2: per-node attention logits ----------------
// One wave per node. Chunk c (32 feats) lies entirely inside head c/2,
// and a_l/a_r flatten to the same 256-feature order as h.
__global__ void __launch_bounds__(256)
gat_node_logits(const float* __restrict__ h, const float* __restrict__ a_l,
                const float* __restrict__ a_r, float* __restrict__ hl,
                float* __restrict__ hr, int N) {
  const int wid = blockIdx.x * 8 + (threadIdx.x >> 5);
  if (wid >= N) return;
  const int lane = threadIdx.x & 31;
  const float* __restrict__ hp = h + (size_t)wid * TOTF;

  float sl[NHEAD] = {0.f, 0.f, 0.f, 0.f};
  float sr[NHEAD] = {0.f, 0.f, 0.f, 0.f};
#pragma unroll
  for (int c = 0; c < 8; ++c) {
    const int f = c * 32 + lane;
    const float hv = hp[f];
    sl[c >> 1] += a_l[f] * hv;
    sr[c >> 1] += a_r[f] * hv;
  }
#pragma unroll
  for (int hd = 0; hd < NHEAD; ++hd) {
    float vl = sl[hd], vr = sr[hd];
#pragma unroll
    for (int off = 16; off > 0; off >>= 1) {
      vl += __shfl_xor(vl, off, 32);
      vr += __shfl_xor(vr, off, 32);
    }
    if (lane == 0) {
      hl[wid * NHEAD + hd] = vl;
      hr[wid * NHEAD + hd] = vr;
    }
  }
}

// ---------------- K3: edge logits + segment max ----------------
__global__ void gat_edge_logits(const int* __restrict__ erow, const int* __restrict__ ecol,
                                const float* __restrict__ hl, const float* __restrict__ hr,
                                float* __restrict__ ebuf, float* __restrict__ emax, int E4) {
  const int i = blockIdx.x * blockDim.x + threadIdx.x;
  if (i >= E4) return;
  const int e = i >> 2, hd = i & 3;
  const int r = erow[e], c = ecol[e];
  float v = hl[r * NHEAD + hd] + hr[c * NHEAD + hd];
  v = (v > 0.0f) ? v : ALPHA * v;             // LeakyReLU
  ebuf[i] = v;
  atomicMaxF32(&emax[r * NHEAD + hd], v);
}

// ---------------- K4: exp + segment sum ----------------
__global__ void gat_edge_exp(const int* __restrict__ erow, const float* __restrict__ emax,
                             float* __restrict__ ebuf, float* __restrict__ denom, int E4) {
  const int i = blockIdx.x * blockDim.x + threadIdx.x;
  if (i >= E4) return;
  const int e = i >> 2, hd = i & 3;
  const int r = erow[e];
  const float ex = __expf(ebuf[i] - emax[r * NHEAD + hd]);
  ebuf[i] = ex;
  atomicAdd(&denom[r * NHEAD + hd], ex);
}

// ---------------- K5: weighted scatter-aggregation ----------------
// One wave per edge; 8 chunks of 32 features; chunk c belongs to head c/2.
__global__ void __launch_bounds__(256)
gat_aggregate(const int* __restrict__ erow, const int* __restrict__ ecol,
              const float* __restrict__ ebuf, const float* __restrict__ denom,
              const float* __restrict__ h, float* __restrict__ out, int E) {
  const int e = blockIdx.x * 8 + (threadIdx.x >> 5);
  if (e >= E) return;
  const int lane = threadIdx.x & 31;
  const int r = erow[e], c = ecol[e];

  float att[NHEAD];
#pragma unroll
  for (int hd = 0; hd < NHEAD; ++hd)
    att[hd] = ebuf[e * NHEAD + hd] / denom[r * NHEAD + hd];

  const float* __restrict__ hc = h + (size_t)c * TOTF;
  float* __restrict__ orow = out + (size_t)r * TOTF;
#pragma unroll
  for (int cc = 0; cc < 8; ++cc) {
    const int f = cc * 32 + lane;
    atomicAdd(&orow[f], att[cc >> 1] * hc[f]);
  }
}

// ---------------- launcher ----------------
extern "C" void kernel_launch(void* const* d_in, const int* in_sizes, int n_in,
                              void* d_out, int out_size, void* d_ws, size_t ws_size,
                              hipStream_t stream) {
  const float* x    = (const float*)d_in[0];
  const int*   edge = (const int*)  d_in[1];
  const float* W    = (const float*)d_in[2];
  const float* a_l  = (const float*)d_in[3];
  const float* a_r  = (const float*)d_in[4];
  float* out = (float*)d_out;

  const int N = in_sizes[0] / IN_F;   // 50000
  const int E = in_sizes[1] / 2;      // 800000
  const int* erow = edge;             // edge[0,:]
  const int* ecol = edge + E;         // edge[1,:]

  // workspace carve-out (floats): h | hl | hr | emax | denom | ebuf  (~67 MB)
  float* ws    = (float*)d_ws;
  float* h     = ws;
  float* hl    = h     + (size_t)N * TOTF;
  float* hr    = hl    + (size_t)N * NHEAD;
  float* emax  = hr    + (size_t)N * NHEAD;
  float* denom = emax  + (size_t)N * NHEAD;
  float* ebuf  = denom + (size_t)N * NHEAD;

  const int n_out = N * TOTF;
  const int n_seg = N * NHEAD;
  const int E4    = E * NHEAD;

  gat_init<<<(n_out + 255) / 256, 256, 0, stream>>>(out, emax, denom, n_out, n_seg);
  gat_gemm_wmma<<<(N + 15) / 16, 128, 0, stream>>>(x, W, h, N);
  gat_node_logits<<<(N + 7) / 8, 256, 0, stream>>>(h, a_l, a_r, hl, hr, N);
  gat_edge_logits<<<(E4 + 255) / 256, 256, 0, stream>>>(erow, ecol, hl, hr, ebuf, emax, E4);
  gat_edge_exp<<<(E4 + 255) / 256, 256, 0, stream>>>(erow, emax, ebuf, denom, E4);
  gat_aggregate<<<(E + 7) / 8, 256, 0, stream>>>(erow, ecol, ebuf, denom, h, out, E);
}